// LaplacianLoss_25434796327108
// MI455X (gfx1250) — compile-verified
//
#include <hip/hip_runtime.h>

// ---------------------------------------------------------------------------
// LaplacianLoss on gfx1250 (MI455X, wave32).
//  kernel 1: fused gather + per-vertex loss, per-block partials -> d_ws
//           (branchless padding mask so all 18 neighbor gathers are in flight
//            simultaneously instead of one serialized wait per neighbor)
//  kernel 2: deterministic reduction of 2048 partials -> d_out[0]
//  Wave reduction: V_WMMA_F32_16X16X4_F32 with all-ones B gives
//  D[m][n] = p[m] + p[m+16]; sum(c[0..7]) is a half-wave sum, one
//  ds_swizzle SWAPX16 exchange completes the 32-lane sum.
// ---------------------------------------------------------------------------

typedef __attribute__((ext_vector_type(2))) float v2f;
typedef __attribute__((ext_vector_type(8))) float v8f;

#define NBLOCKS  2048
#define NTHREADS 256
#define DEG      9

struct F3 { float x, y, z; };   // 12 bytes, align 4 -> global_load_b96

__device__ __forceinline__ float swap_x16(float x) {
    // ds_swizzle_b32 group-of-32: offset[14:10]=xor=0x10, or=0, and=0x1f
    int r = __builtin_amdgcn_ds_swizzle(__float_as_int(x), 0x401F);
    return __int_as_float(r);
}

__device__ __forceinline__ float wave_reduce_f32(float v) {
    // A: 16x4 f32. Lanes 0-15 vgpr0 = (M=lane,K=0); lanes 16-31 vgpr0 = (M=lane-16,K=2).
    v2f a; a.x = v;    a.y = 0.0f;
    v2f b; b.x = 1.0f; b.y = 1.0f;   // B: 4x16 ones
    v8f c = {};
    c = __builtin_amdgcn_wmma_f32_16x16x4_f32(
            /*neg_a=*/false, a, /*neg_b=*/false, b,
            /*c_mod=*/(short)0, c, /*reuse_a=*/false, /*reuse_b=*/false);
    // lane n (n<16): sum_m D[0..7][n]  = sum(p[0..7])  + sum(p[16..23])
    // lane n+16:     sum_m D[8..15][n] = sum(p[8..15]) + sum(p[24..31])
    float s = c[0] + c[1] + c[2] + c[3] + c[4] + c[5] + c[6] + c[7];
    s += swap_x16(s);                // combine half-sums -> full wave sum
    return s;
}

__global__ void __launch_bounds__(NTHREADS)
lap_loss_partial(const float* __restrict__ v1, const float* __restrict__ v2,
                 const int*   __restrict__ adj,
                 const float* __restrict__ w,  const float* __restrict__ lw,
                 float* __restrict__ partial, int n) {
    const int tid  = threadIdx.x;
    const int gtid = blockIdx.x * NTHREADS + tid;
    const F3* V1 = (const F3*)v1;
    const F3* V2 = (const F3*)v2;

    float acc = 0.0f;
    for (int i = gtid; i < n; i += NBLOCKS * NTHREADS) {
        const int base = i * DEG;
        // prefetch next grid-stride iteration's adjacency row
        const int nexti = i + NBLOCKS * NTHREADS;
        if (nexti < n) __builtin_prefetch(&adj[nexti * DEG], 0, 0);

        // 1) load all 9 indices, build branchless masks (index==n is padding)
        int   ks[DEG];
        float ms[DEG];
#pragma unroll
        for (int j = 0; j < DEG; ++j) {
            const int k   = adj[base + j];
            const bool ok = ((unsigned)k < (unsigned)n);
            ms[j] = ok ? 1.0f : 0.0f;
            ks[j] = ok ? k    : 0;
        }

        // 2) issue all 18 gathers unconditionally; accumulate with mask FMAs
        float sx = 0.0f, sy = 0.0f, sz = 0.0f;
#pragma unroll
        for (int j = 0; j < DEG; ++j) {
            const F3 a = V1[ks[j]];
            const F3 b = V2[ks[j]];
            sx = __builtin_fmaf(ms[j], a.x - b.x, sx);
            sy = __builtin_fmaf(ms[j], a.y - b.y, sy);
            sz = __builtin_fmaf(ms[j], a.z - b.z, sz);
        }

        const float invw = 1.0f / w[i];
        const F3 a = V1[i];
        const F3 b = V2[i];
        const float dx = (a.x - b.x) - sx * invw;
        const float dy = (a.y - b.y) - sy * invw;
        const float dz = (a.z - b.z) - sz * invw;
        acc += lw[i] * (dx * dx + dy * dy + dz * dz);
    }

    // all lanes reconverged here; EXEC is all-1s for WMMA
    const float s = wave_reduce_f32(acc);

    __shared__ float wsum[NTHREADS / 32];
    const int lane = tid & 31;
    const int wv   = tid >> 5;
    if (lane == 0) wsum[wv] = s;
    __syncthreads();
    if (tid == 0) {
        float t = 0.0f;
#pragma unroll
        for (int k = 0; k < NTHREADS / 32; ++k) t += wsum[k];
        partial[blockIdx.x] = t;         // fixed order -> deterministic
    }
}

__global__ void __launch_bounds__(NTHREADS)
lap_loss_final(const float* __restrict__ partial, float* __restrict__ out,
               float inv_count) {
    const int tid = threadIdx.x;
    float acc = 0.0f;
    for (int i = tid; i < NBLOCKS; i += NTHREADS) acc += partial[i];

    const float s = wave_reduce_f32(acc);

    __shared__ float wsum[NTHREADS / 32];
    const int lane = tid & 31;
    const int wv   = tid >> 5;
    if (lane == 0) wsum[wv] = s;
    __syncthreads();
    if (tid == 0) {
        float t = 0.0f;
#pragma unroll
        for (int k = 0; k < NTHREADS / 32; ++k) t += wsum[k];
        out[0] = t * inv_count;
    }
}

extern "C" void kernel_launch(void* const* d_in, const int* in_sizes, int n_in,
                              void* d_out, int out_size, void* d_ws, size_t ws_size,
                              hipStream_t stream) {
    const float* v1  = (const float*)d_in[0];
    const float* v2  = (const float*)d_in[1];
    const int*   adj = (const int*)  d_in[2];
    const float* w   = (const float*)d_in[3];
    const float* lw  = (const float*)d_in[4];

    const int n = in_sizes[0] / 3;           // N vertices
    float* partial = (float*)d_ws;           // NBLOCKS floats of scratch

    lap_loss_partial<<<NBLOCKS, NTHREADS, 0, stream>>>(v1, v2, adj, w, lw, partial, n);

    const float inv_count = 1.0f / (3.0f * (float)n);
    lap_loss_final<<<1, NTHREADS, 0, stream>>>(partial, (float*)d_out, inv_count);
}